// DiffusionActionHead_32865089749179
// MI455X (gfx1250) — compile-verified
//
#include <hip/hip_runtime.h>
#include <math.h>

#define BATCH 8
#define SEQ   2048
#define DIM   4096
#define NHEAD 8
#define DHEAD 512
#define HID   256
#define KCAT  4136   // 32 + 4096 + 7, padded to multiple of 4

typedef __attribute__((ext_vector_type(2))) float v2f;
typedef __attribute__((ext_vector_type(8))) float v8f;

__device__ __forceinline__ v8f wmma4(v2f a, v2f b, v8f c) {
    return __builtin_amdgcn_wmma_f32_16x16x4_f32(false, a, false, b, (short)0, c, false, false);
}

// ---------------------------------------------------------------------------
// Fused 16xK @ KxN WMMA GEMM (small GEMMs). One wave per 16-col tile.
// ACT: 0=none, 1=exact GELU, 2=swish.  RES: add residual at store.
// ---------------------------------------------------------------------------
template<int ACT, bool RES>
__global__ void __launch_bounds__(32)
gemm16_wmma(const float* __restrict__ A, int lda,
            const float* __restrict__ W, int ldw,
            const float* __restrict__ bias,
            const float* __restrict__ Rres, int ldr,
            float* __restrict__ C, int ldc, int K)
{
    const int lane = threadIdx.x;
    const int n0   = blockIdx.x * 16;
    const int m    = lane & 15;
    const int koff = (lane >> 4) * 2;

    const float* ap = A + (size_t)m * lda + koff;
    const float* wp = W + (size_t)koff * ldw + n0 + m;

    v8f acc = {};
    #pragma unroll 4
    for (int k = 0; k < K; k += 4) {
        v2f a; a.x = ap[0];   a.y = ap[1];
        v2f b; b.x = wp[0];   b.y = wp[ldw];
        acc = wmma4(a, b, acc);
        ap += 4;
        wp += (size_t)4 * ldw;
    }

    const int col = n0 + m;
    const float bv = bias[col];
    #pragma unroll
    for (int r = 0; r < 8; ++r) {
        const int row = r + ((lane >> 4) << 3);
        float v = acc[r] + bv;
        if (ACT == 1) v = 0.5f * v * (1.0f + erff(v * 0.70710678118654752f));
        else if (ACT == 2) v = v / (1.0f + __expf(-v));
        if (RES) v += Rres[(size_t)row * ldr + col];
        C[(size_t)row * ldc + col] = v;
    }
}

// ---------------------------------------------------------------------------
// K-split WMMA GEMM for the big weight streams. One wave per 32-col tile and
// per K-slice; writes raw partials P[ks][16][N]. aShift groups A per column
// block (ctx per-head A switching): A += (n0 >> aShift) * 16 * lda.
// ---------------------------------------------------------------------------
__global__ void __launch_bounds__(32)
gemm16_wmma_split(const float* __restrict__ A, int lda, int aShift,
                  const float* __restrict__ W, int ldw,
                  float* __restrict__ P, int N, int Kslice)
{
    const int lane = threadIdx.x;
    const int n0   = blockIdx.x * 32;
    const int ks   = blockIdx.y;
    const int m    = lane & 15;
    const int koff = (lane >> 4) * 2;
    const int k0   = ks * Kslice;

    const float* ap  = A + (size_t)(n0 >> aShift) * 16 * lda + (size_t)m * lda + k0 + koff;
    const float* wp0 = W + (size_t)(k0 + koff) * ldw + n0 + m;
    const float* wp1 = wp0 + 16;

    v8f acc0 = {}, acc1 = {};
    #pragma unroll 4
    for (int k = 0; k < Kslice; k += 4) {
        v2f a;  a.x  = ap[0];   a.y  = ap[1];
        v2f b0; b0.x = wp0[0];  b0.y = wp0[ldw];
        v2f b1; b1.x = wp1[0];  b1.y = wp1[ldw];
        acc0 = wmma4(a, b0, acc0);
        acc1 = wmma4(a, b1, acc1);
        ap  += 4;
        wp0 += (size_t)4 * ldw;
        wp1 += (size_t)4 * ldw;
    }

    const int col = n0 + m;
    float* pb = P + (size_t)ks * 16 * N;
    #pragma unroll
    for (int r = 0; r < 8; ++r) {
        const int row = r + ((lane >> 4) << 3);
        pb[(size_t)row * N + col]      = acc0[r];
        pb[(size_t)row * N + col + 16] = acc1[r];
    }
}

// --- combine K-split partials, apply bias / activation / residual ----------
template<int ACT, bool RES>
__global__ void combine16(const float* __restrict__ P, int nsplit,
                          const float* __restrict__ bias,
                          const float* __restrict__ Rres, int ldr,
                          float* __restrict__ C, int ldc, int N)
{
    const int idx = blockIdx.x * 256 + threadIdx.x;
    if (idx >= 16 * N) return;
    const int row = idx / N, col = idx % N;
    float v = 0.f;
    for (int s = 0; s < nsplit; ++s)
        v += P[((size_t)s * 16 + row) * N + col];
    v += bias[col];
    if (ACT == 1) v = 0.5f * v * (1.0f + erff(v * 0.70710678118654752f));
    else if (ACT == 2) v = v / (1.0f + __expf(-v));
    if (RES) v += Rres[(size_t)row * ldr + col];
    C[(size_t)row * ldc + col] = v;
}

// --- q partial projection: qpart[ic][j] = sum_{i in chunk} probe[i]*wq[i,j] ---
__global__ void qproj_kernel(const float* __restrict__ probe,
                             const float* __restrict__ wq,
                             float* __restrict__ qpart)
{
    __shared__ float sp[256];
    const int jc = blockIdx.x & 15, ic = blockIdx.x >> 4;
    const int tid = threadIdx.x;
    sp[tid] = probe[ic * 256 + tid];
    __syncthreads();
    const int j = jc * 256 + tid;
    const float* wp = wq + (size_t)(ic * 256) * DIM + j;
    float acc = 0.f;
    #pragma unroll 8
    for (int ii = 0; ii < 256; ++ii) acc = fmaf(sp[ii], wp[(size_t)ii * DIM], acc);
    qpart[ic * DIM + j] = acc;
}

__global__ void qred_kernel(const float* __restrict__ qpart,
                            const float* __restrict__ bq,
                            float* __restrict__ q)
{
    const int j = blockIdx.x * 256 + threadIdx.x;
    float s = bq[j];
    #pragma unroll
    for (int ic = 0; ic < 16; ++ic) s += qpart[ic * DIM + j];
    q[j] = s;
}

// --- sbias[h] = sum_{j in head h} bk[j]*q[j] -------------------------------
__global__ void sbias_kernel(const float* __restrict__ bk,
                             const float* __restrict__ q,
                             float* __restrict__ sbias)
{
    const int lane = threadIdx.x & 31, h = threadIdx.x >> 5;
    float p = 0.f;
    for (int t = 0; t < 16; ++t) {
        const int j = h * DHEAD + lane + 32 * t;
        p = fmaf(bk[j], q[j], p);
    }
    for (int o = 16; o; o >>= 1) p += __shfl_xor(p, o, 32);
    if (lane == 0) sbias[h] = p;
}

// --- w2[h][i] = sum_{j in head h} wk[i,j]*q[j] (one wave per row i) --------
__global__ void w2_kernel(const float* __restrict__ wk,
                          const float* __restrict__ q,
                          float* __restrict__ w2)
{
    const int lane = threadIdx.x & 31;
    const int i = blockIdx.x * 8 + (threadIdx.x >> 5);
    const float* row = wk + (size_t)i * DIM;
    #pragma unroll
    for (int h = 0; h < NHEAD; ++h) {
        float p = 0.f;
        const float* rh = row + h * DHEAD;
        const float* qh = q + h * DHEAD;
        for (int t = 0; t < 16; ++t)
            p = fmaf(rh[lane + 32 * t], qh[lane + 32 * t], p);
        for (int o = 16; o; o >>= 1) p += __shfl_xor(p, o, 32);
        if (lane == 0) w2[h * DIM + i] = p;
    }
}

// --- scores[b,h,s] = (llm[b,s]·w2[h] + sbias[h]) / sqrt(DH) (wave per (b,s))
__global__ void scores_kernel(const float* __restrict__ llm,
                              const float* __restrict__ w2,
                              const float* __restrict__ sbias,
                              float* __restrict__ sc)
{
    const int lane = threadIdx.x & 31;
    const int gw = blockIdx.x * 8 + (threadIdx.x >> 5);
    const int b = gw >> 11, s = gw & (SEQ - 1);
    const float* row = llm + ((size_t)b * SEQ + s) * DIM;
    float acc[NHEAD] = {};
    for (int t = 0; t < DIM / 32; ++t) {
        const float x = row[lane + 32 * t];
        #pragma unroll
        for (int h = 0; h < NHEAD; ++h)
            acc[h] = fmaf(x, w2[h * DIM + lane + 32 * t], acc[h]);
    }
    #pragma unroll
    for (int h = 0; h < NHEAD; ++h)
        for (int o = 16; o; o >>= 1) acc[h] += __shfl_xor(acc[h], o, 32);
    if (lane == 0) {
        const float scale = 0.04419417382415922f; // 1/sqrt(512)
        #pragma unroll
        for (int h = 0; h < NHEAD; ++h)
            sc[((size_t)(b * NHEAD + h)) * SEQ + s] = (acc[h] + sbias[h]) * scale;
    }
}

// --- softmax over S per (b,h) ----------------------------------------------
__global__ void softmax_kernel(float* __restrict__ sc)
{
    __shared__ float red[256];
    float* p = sc + (size_t)blockIdx.x * SEQ;
    const int tid = threadIdx.x;
    float m = -3.4e38f;
    for (int j = tid; j < SEQ; j += 256) m = fmaxf(m, p[j]);
    red[tid] = m; __syncthreads();
    for (int o = 128; o; o >>= 1) { if (tid < o) red[tid] = fmaxf(red[tid], red[tid + o]); __syncthreads(); }
    m = red[0]; __syncthreads();
    float s = 0.f;
    for (int j = tid; j < SEQ; j += 256) { const float e = __expf(p[j] - m); p[j] = e; s += e; }
    red[tid] = s; __syncthreads();
    for (int o = 128; o; o >>= 1) { if (tid < o) red[tid] += red[tid + o]; __syncthreads(); }
    const float inv = 1.0f / red[0];
    for (int j = tid; j < SEQ; j += 256) p[j] *= inv;
}

// --- pool partials: pp[c][b,h,i] = sum_{s in chunk c} attn[b,h,s]*llm[b,s,i]
__global__ void pool_kernel(const float* __restrict__ llm,
                            const float* __restrict__ attn,
                            float* __restrict__ pp)
{
    __shared__ float sa[NHEAD][256];
    const int ic = blockIdx.x & 15;
    const int c  = (blockIdx.x >> 4) & 7;
    const int b  = blockIdx.x >> 7;
    const int tid = threadIdx.x;
    #pragma unroll
    for (int h = 0; h < NHEAD; ++h)
        sa[h][tid] = attn[((size_t)(b * NHEAD + h)) * SEQ + c * 256 + tid];
    __syncthreads();
    const int i = ic * 256 + tid;
    float acc[NHEAD] = {};
    const float* base = llm + ((size_t)b * SEQ + c * 256) * DIM + i;
    for (int sp = 0; sp < 256; ++sp) {
        const float x = base[(size_t)sp * DIM];
        #pragma unroll
        for (int h = 0; h < NHEAD; ++h) acc[h] = fmaf(sa[h][sp], x, acc[h]);
    }
    float* o = pp + (((size_t)(c * BATCH + b)) * NHEAD) * DIM + i;
    #pragma unroll
    for (int h = 0; h < NHEAD; ++h) o[(size_t)h * DIM] = acc[h];
}

// --- reduce partials into per-head padded A matrices pool16[h] (16 x DIM) --
__global__ void poolred_kernel(const float* __restrict__ pp,
                               float* __restrict__ pool16)
{
    const int idx = blockIdx.x * 256 + threadIdx.x;   // h*DIM + i
    const int h = idx / DIM, i = idx % DIM;
    #pragma unroll
    for (int b = 0; b < BATCH; ++b) {
        float s = 0.f;
        #pragma unroll
        for (int c = 0; c < 8; ++c)
            s += pp[(((size_t)(c * BATCH + b)) * NHEAD + h) * DIM + i];
        pool16[((size_t)h * 16 + b) * DIM + i] = s;
    }
    #pragma unroll
    for (int b = BATCH; b < 16; ++b) pool16[((size_t)h * 16 + b) * DIM + i] = 0.f;
}

// --- layernorm on rows 0..7 of a padded 16xN matrix; rows >=8 zeroed -------
__global__ void ln16_kernel(const float* __restrict__ X,
                            const float* __restrict__ g,
                            const float* __restrict__ bb,
                            float* __restrict__ Y, int N)
{
    __shared__ float red[256];
    const int row = blockIdx.x, tid = threadIdx.x;
    float* y = Y + (size_t)row * N;
    if (row >= 8) { for (int j = tid; j < N; j += 256) y[j] = 0.f; return; }
    const float* x = X + (size_t)row * N;
    float s = 0.f, ss = 0.f;
    for (int j = tid; j < N; j += 256) { const float v = x[j]; s += v; ss += v * v; }
    red[tid] = s; __syncthreads();
    for (int o = 128; o; o >>= 1) { if (tid < o) red[tid] += red[tid + o]; __syncthreads(); }
    const float mean = red[0] / (float)N; __syncthreads();
    red[tid] = ss; __syncthreads();
    for (int o = 128; o; o >>= 1) { if (tid < o) red[tid] += red[tid + o]; __syncthreads(); }
    const float var = red[0] / (float)N - mean * mean;
    const float inv = rsqrtf(var + 1e-5f);
    for (int j = tid; j < N; j += 256) y[j] = (x[j] - mean) * inv * g[j] + bb[j];
}

// --- fourier features + cond MLP (tiny) ------------------------------------
__global__ void cond_kernel(const float* __restrict__ time,
                            const float* __restrict__ four_w,
                            const float* __restrict__ w1, const float* __restrict__ b1,
                            const float* __restrict__ w2, const float* __restrict__ b2,
                            float* __restrict__ cond)
{
    const int b = threadIdx.x;
    if (b >= BATCH) return;
    const float t = time[b];
    float ff[32];
    for (int j = 0; j < 16; ++j) {
        const float f = 6.283185307179586f * t * four_w[j];
        ff[j] = cosf(f); ff[16 + j] = sinf(f);
    }
    float h[64];
    for (int o = 0; o < 64; ++o) {
        float a = b1[o];
        for (int i = 0; i < 32; ++i) a = fmaf(ff[i], w1[i * 64 + o], a);
        h[o] = a / (1.0f + __expf(-a));
    }
    for (int o = 0; o < 32; ++o) {
        float a = b2[o];
        for (int i = 0; i < 64; ++i) a = fmaf(h[i], w2[i * 32 + o], a);
        cond[b * 32 + o] = a;
    }
}

// --- build padded concat (16 x KCAT): [cond | pooled | noisy | 0pad] -------
__global__ void concat_kernel(const float* __restrict__ cond,
                              const float* __restrict__ pooled16,
                              const float* __restrict__ noisy,
                              float* __restrict__ out)
{
    const int idx = blockIdx.x * 256 + threadIdx.x;
    if (idx >= 16 * KCAT) return;
    const int row = idx / KCAT, col = idx % KCAT;
    float v = 0.f;
    if (row < 8) {
        if (col < 32)            v = cond[row * 32 + col];
        else if (col < 32 + DIM) v = pooled16[(size_t)row * DIM + (col - 32)];
        else if (col < 4135)     v = noisy[row * 7 + (col - 4128)];
    }
    out[idx] = v;
}

// --- pad rin_w (4135x256) -> (4136x256) ------------------------------------
__global__ void padrinw_kernel(const float* __restrict__ w, float* __restrict__ o)
{
    const int idx = blockIdx.x * 256 + threadIdx.x;
    o[idx] = ((idx >> 8) < 4135) ? w[idx] : 0.f;
}

// --- final: out = swish(x) @ out_w + out_b ---------------------------------
__global__ void out_kernel(const float* __restrict__ x,
                           const float* __restrict__ w,
                           const float* __restrict__ b,
                           float* __restrict__ out)
{
    const int t = threadIdx.x;
    if (t >= BATCH * 7) return;
    const int bb = t / 7, a = t % 7;
    float acc = b[a];
    for (int k = 0; k < HID; ++k) {
        float v = x[bb * HID + k];
        v = v / (1.0f + __expf(-v));
        acc = fmaf(v, w[k * 7 + a], acc);
    }
    out[t] = acc;
}

// ---------------------------------------------------------------------------
extern "C" void kernel_launch(void* const* d_in, const int* in_sizes, int n_in,
                              void* d_out, int out_size, void* d_ws, size_t ws_size,
                              hipStream_t stream)
{
    (void)in_sizes; (void)n_in; (void)out_size; (void)ws_size;
    const float* llm    = (const float*)d_in[0];
    const float* timei  = (const float*)d_in[1];
    const float* noisy  = (const float*)d_in[2];
    const float* probe  = (const float*)d_in[3];
    const float* wq     = (const float*)d_in[4];
    const float* wk     = (const float*)d_in[5];
    const float* wv     = (const float*)d_in[6];
    const float* bq     = (const float*)d_in[7];
    const float* bk     = (const float*)d_in[8];
    const float* bv     = (const float*)d_in[9];
    const float* wo     = (const float*)d_in[10];
    const float* bo     = (const float*)d_in[11];
    const float* ln_g   = (const float*)d_in[12];
    const float* ln_b   = (const float*)d_in[13];
    const float* mw1    = (const float*)d_in[14];
    const float* mb1    = (const float*)d_in[15];
    const float* mw2    = (const float*)d_in[16];
    const float* mb2    = (const float*)d_in[17];
    const float* four_w = (const float*)d_in[18];
    const float* cw1    = (const float*)d_in[19];
    const float* cb1    = (const float*)d_in[20];
    const float* cw2    = (const float*)d_in[21];
    const float* cb2    = (const float*)d_in[22];
    const float* rin_w  = (const float*)d_in[23];
    const float* rin_b  = (const float*)d_in[24];
    const float* blg    = (const float*)d_in[25];
    const float* blb    = (const float*)d_in[26];
    const float* bw1    = (const float*)d_in[27];
    const float* bb1    = (const float*)d_in[28];
    const float* bw2    = (const float*)d_in[29];
    const float* bb2    = (const float*)d_in[30];
    const float* ow     = (const float*)d_in[31];
    const float* ob     = (const float*)d_in[32];
    float* out = (float*)d_out;

    // workspace carve-up (floats)
    float* ws = (float*)d_ws;
    float* q        = ws;                 ws += DIM;
    float* qpart    = ws;                 ws += 16 * DIM;
    float* sbias    = ws;                 ws += 32;
    float* w2       = ws;                 ws += NHEAD * DIM;
    float* sc       = ws;                 ws += BATCH * NHEAD * SEQ;
    float* pp       = ws;                 ws += 8 * BATCH * NHEAD * DIM;    // 2M
    float* pool16   = ws;                 ws += NHEAD * 16 * DIM;           // 512K
    float* ctx16    = ws;                 ws += 16 * DIM;
    float* ao16     = ws;                 ws += 16 * DIM;
    float* y16      = ws;                 ws += 16 * DIM;
    float* h16      = ws;                 ws += 16 * 4 * DIM;               // 256K
    float* pooled16 = ws;                 ws += 16 * DIM;
    float* gpart    = ws;                 ws += 8 * 16 * DIM;               // 512K (max split partials)
    float* cond     = ws;                 ws += 256;
    float* cc16     = ws;                 ws += 16 * KCAT;
    float* rinwp    = ws;                 ws += KCAT * HID;
    float* xa       = ws;                 ws += 16 * HID;
    float* xb       = ws;                 ws += 16 * HID;
    float* lnb16    = ws;                 ws += 16 * HID;
    float* h1_16    = ws;                 ws += 16 * 4 * HID;

    // --- MAPHead, algebraically collapsed ---
    qproj_kernel<<<256, 256, 0, stream>>>(probe, wq, qpart);
    qred_kernel<<<16, 256, 0, stream>>>(qpart, bq, q);
    sbias_kernel<<<1, 256, 0, stream>>>(bk, q, sbias);
    w2_kernel<<<DIM / 8, 256, 0, stream>>>(wk, q, w2);
    scores_kernel<<<BATCH * SEQ / 8, 256, 0, stream>>>(llm, w2, sbias, sc);
    softmax_kernel<<<BATCH * NHEAD, 256, 0, stream>>>(sc);
    pool_kernel<<<BATCH * 8 * (DIM / 256), 256, 0, stream>>>(llm, sc, pp);
    poolred_kernel<<<NHEAD * DIM / 256, 256, 0, stream>>>(pp, pool16);

    // ctx = pool @ wv + bv  (grouped per-head A via aShift=9; K-split 4)
    gemm16_wmma_split<<<dim3(DIM / 32, 4), 32, 0, stream>>>(
        pool16, DIM, 9, wv, DIM, gpart, DIM, DIM / 4);
    combine16<0, false><<<(16 * DIM + 255) / 256, 256, 0, stream>>>(
        gpart, 4, bv, nullptr, 0, ctx16, DIM, DIM);

    // attn_out = ctx @ wo + bo  (K-split 4)
    gemm16_wmma_split<<<dim3(DIM / 32, 4), 32, 0, stream>>>(
        ctx16, DIM, 31, wo, DIM, gpart, DIM, DIM / 4);
    combine16<0, false><<<(16 * DIM + 255) / 256, 256, 0, stream>>>(
        gpart, 4, bo, nullptr, 0, ao16, DIM, DIM);

    // y = LN(attn_out)
    ln16_kernel<<<16, 256, 0, stream>>>(ao16, ln_g, ln_b, y16, DIM);

    // h = gelu(y @ mlp_w1 + b1)  (N=16384, K-split 2)
    gemm16_wmma_split<<<dim3(4 * DIM / 32, 2), 32, 0, stream>>>(
        y16, DIM, 31, mw1, 4 * DIM, gpart, 4 * DIM, DIM / 2);
    combine16<1, false><<<(16 * 4 * DIM + 255) / 256, 256, 0, stream>>>(
        gpart, 2, mb1, nullptr, 0, h16, 4 * DIM, 4 * DIM);

    // pooled = attn_out + h @ mlp_w2 + b2  (K=16384, K-split 8)
    gemm16_wmma_split<<<dim3(DIM / 32, 8), 32, 0, stream>>>(
        h16, 4 * DIM, 31, mw2, DIM, gpart, DIM, 4 * DIM / 8);
    combine16<0, true><<<(16 * DIM + 255) / 256, 256, 0, stream>>>(
        gpart, 8, mb2, ao16, DIM, pooled16, DIM, DIM);

    // --- diffusion head ---
    cond_kernel<<<1, 32, 0, stream>>>(timei, four_w, cw1, cb1, cw2, cb2, cond);
    padrinw_kernel<<<KCAT, 256, 0, stream>>>(rin_w, rinwp);
    concat_kernel<<<(16 * KCAT + 255) / 256, 256, 0, stream>>>(cond, pooled16, noisy, cc16);
    gemm16_wmma<0, false><<<HID / 16, 32, 0, stream>>>(
        cc16, KCAT, rinwp, HID, rin_b, nullptr, 0, xa, HID, KCAT);

    float* xcur = xa;
    float* xnxt = xb;
    for (int i = 0; i < 3; ++i) {
        ln16_kernel<<<16, 256, 0, stream>>>(xcur, blg + i * HID, blb + i * HID, lnb16, HID);
        gemm16_wmma<2, false><<<4 * HID / 16, 32, 0, stream>>>(
            lnb16, HID, bw1 + (size_t)i * HID * 4 * HID, 4 * HID,
            bb1 + i * 4 * HID, nullptr, 0, h1_16, 4 * HID, HID);
        gemm16_wmma<0, true><<<HID / 16, 32, 0, stream>>>(
            h1_16, 4 * HID, bw2 + (size_t)i * 4 * HID * HID, HID,
            bb2 + i * HID, xcur, HID, xnxt, HID, 4 * HID);
        float* t = xcur; xcur = xnxt; xnxt = t;
    }
    out_kernel<<<1, 64, 0, stream>>>(xcur, ow, ob, out);
}